// RGCN_65687229825992
// MI455X (gfx1250) — compile-verified
//
#include <hip/hip_runtime.h>

// RGCN 2-layer forward for MI455X (gfx1250, wave32).
//
// Edges are bucketed ONCE by key = dst*16 + type (histogram -> 3-level scan ->
// placement). Each layer is then a single fused kernel: per 16-node tile,
// per relation, gather+sum contiguous source rows (registers, no atomics),
// stage in LDS (padded rows -> conflict-free ds_load_b64 A-fragments), and
// accumulate via v_wmma_f32_16x16x4_f32. Root GEMM + 1/deg scaling + ReLU are
// fused into the same kernel. fp32 end-to-end (the problem is bandwidth-bound,
// so the slow-but-exact f32 WMMA rate is free; no precision loss).

typedef __attribute__((ext_vector_type(2))) float v2f;
typedef __attribute__((ext_vector_type(8))) float v8f;

#define NDIM 128
#define RNUM 16
#define LROW 132            // LDS row stride in floats (128 + 4 pad): banks 4m+k

__global__ void zero_i4(int4* __restrict__ p, long n4) {
  long i = (long)blockIdx.x * blockDim.x + threadIdx.x;
  if (i < n4) p[i] = make_int4(0, 0, 0, 0);
}

// X[row,:] = emb[entity[row],:]  -- one thread per 4 floats (32 threads/row)
__global__ void gather_rows(const int* __restrict__ entity,
                            const float* __restrict__ emb,
                            float* __restrict__ X, int nRows) {
  int t = blockIdx.x * blockDim.x + threadIdx.x;
  int row = t >> 5;
  int q   = t & 31;
  if (row >= nRows) return;
  const float4* s = (const float4*)(emb + (size_t)entity[row] * NDIM);
  float4* d = (float4*)(X + (size_t)row * NDIM);
  d[q] = s[q];
}

// Per-(dst,type) bucket histogram. Native u32 atomics.
__global__ void histo(const int* __restrict__ dst, const int* __restrict__ et,
                      int* __restrict__ cnt, int E) {
  int e = blockIdx.x * blockDim.x + threadIdx.x;
  if (e >= E) return;
  atomicAdd(&cnt[dst[e] * RNUM + et[e]], 1);
}

// Block-local exclusive scan (Hillis-Steele in LDS), emits block totals.
__global__ void scan_block(const int* __restrict__ in, int* __restrict__ outExcl,
                           int* __restrict__ blockSums, int n) {
  __shared__ int sh[256];
  int i = blockIdx.x * 256 + threadIdx.x;
  int v = (i < n) ? in[i] : 0;
  sh[threadIdx.x] = v;
  __syncthreads();
  for (int d = 1; d < 256; d <<= 1) {
    int u = (threadIdx.x >= d) ? sh[threadIdx.x - d] : 0;
    __syncthreads();
    sh[threadIdx.x] += u;
    __syncthreads();
  }
  if (i < n) outExcl[i] = sh[threadIdx.x] - v;
  if (threadIdx.x == 255 && blockSums) blockSums[blockIdx.x] = sh[255];
}

__global__ void scan_fix(int* __restrict__ off, const int* __restrict__ s1x,
                         const int* __restrict__ s2x, int n) {
  int i = blockIdx.x * 256 + threadIdx.x;
  if (i < n) off[i] += s1x[i >> 8] + s2x[i >> 16];
}

// srcSorted[off[key] + rank] = src[e]
__global__ void place_edges(const int* __restrict__ src, const int* __restrict__ dst,
                            const int* __restrict__ et, const int* __restrict__ off,
                            int* __restrict__ cursor, int* __restrict__ srcS, int E) {
  int e = blockIdx.x * blockDim.x + threadIdx.x;
  if (e >= E) return;
  int key = dst[e] * RNUM + et[e];
  int pos = off[key] + atomicAdd(&cursor[key], 1);
  srcS[pos] = src[e];
}

// inv[n] = 1 / max(deg(n), 1); deg from the 16 contiguous bucket counts.
__global__ void make_inv(const int* __restrict__ cnt, float* __restrict__ inv, int N) {
  int n_ = blockIdx.x * blockDim.x + threadIdx.x;
  if (n_ >= N) return;
  int d = 0;
#pragma unroll
  for (int r = 0; r < RNUM; r++) d += cnt[n_ * RNUM + r];
  inv[n_] = 1.0f / fmaxf((float)d, 1.0f);
}

// One fused RGCN layer. One wave per 16-row tile; 4 waves / 128-thread block.
// Fragment layouts (wave32, V_WMMA_F32_16X16X4_F32):
//   A 16x4 : lane m = lane&15, half = lane>>4; vgpr j -> K = 2*half + j
//   B 4x16 : lane n = lane&15, half selects K pair; vgpr j -> row K = 2*half + j
//   C 16x16: lane n = lane&15, vgpr v -> row M = (lane>>4)*8 + v
__global__ void __launch_bounds__(128)
rgcn_layer(const float* __restrict__ Xin, const int* __restrict__ srcS,
           const int* __restrict__ off, const int* __restrict__ cnt,
           const float* __restrict__ Wrel, const float* __restrict__ Wroot,
           const float* __restrict__ inv, float* __restrict__ Yout,
           int nRows, int do_relu) {
  __shared__ alignas(16) float lds[4 * 16 * LROW];   // 33,792 B
  int wlocal = threadIdx.x >> 5;
  int wid    = blockIdx.x * 4 + wlocal;
  int row0   = wid << 4;
  if (row0 >= nRows) return;                 // uniform: EXEC stays all-ones
  int lane  = threadIdx.x & 31;
  int n     = lane & 15;
  int half  = lane >> 4;
  int mbase = half << 3;
  float* lp = lds + wlocal * (16 * LROW);

  v8f acc[8];
  const v8f zv = {0.f, 0.f, 0.f, 0.f, 0.f, 0.f, 0.f, 0.f};
#pragma unroll
  for (int t = 0; t < 8; t++) acc[t] = zv;

  for (int r = 0; r < RNUM; r++) {
    // ---- neighbor sums for relation r: registers -> LDS (no atomics) ----
#pragma unroll
    for (int m = 0; m < 16; m++) {
      int key = (row0 + m) * RNUM + r;       // uniform across the wave
      int beg = off[key];
      int c   = cnt[key];
      float4 s = make_float4(0.f, 0.f, 0.f, 0.f);
      for (int j = 0; j < c; j++) {          // avg ~0.6 iterations
        int sn = srcS[beg + j];              // uniform
        float4 v = ((const float4*)(Xin + (size_t)sn * NDIM))[lane];
        s.x += v.x; s.y += v.y; s.z += v.z; s.w += v.w;
      }
      ((float4*)(lp + m * LROW))[lane] = s;  // ds_store_b128, conflict-free
    }
    // ---- acc += S_tile @ W_r ----
    const float* B = Wrel + (size_t)r * NDIM * NDIM;
#pragma unroll 2
    for (int k0 = 0; k0 < NDIM; k0 += 4) {
      int ka = k0 + (half << 1);
      v2f a = *(const v2f*)(lp + n * LROW + ka);   // ds_load_b64, banks 4m+ka
#pragma unroll
      for (int t = 0; t < 8; t++) {
        v2f b;
        b.x = B[(size_t)ka * NDIM + t * 16 + n];
        b.y = B[(size_t)(ka + 1) * NDIM + t * 16 + n];
        acc[t] = __builtin_amdgcn_wmma_f32_16x16x4_f32(
            false, a, false, b, (short)0, acc[t], false, false);
      }
    }
  }

  // ---- mean: scale by 1/deg (per output row) ----
  float iv[8];
#pragma unroll
  for (int v = 0; v < 8; v++) iv[v] = inv[row0 + mbase + v];
#pragma unroll
  for (int t = 0; t < 8; t++)
#pragma unroll
    for (int v = 0; v < 8; v++) acc[t][v] *= iv[v];

  // ---- fused root term: acc += X_tile @ Wroot ----
#pragma unroll
  for (int m = 0; m < 16; m++)
    ((float4*)(lp + m * LROW))[lane] =
        ((const float4*)(Xin + (size_t)(row0 + m) * NDIM))[lane];
#pragma unroll 2
  for (int k0 = 0; k0 < NDIM; k0 += 4) {
    int ka = k0 + (half << 1);
    v2f a = *(const v2f*)(lp + n * LROW + ka);
#pragma unroll
    for (int t = 0; t < 8; t++) {
      v2f b;
      b.x = Wroot[(size_t)ka * NDIM + t * 16 + n];
      b.y = Wroot[(size_t)(ka + 1) * NDIM + t * 16 + n];
      acc[t] = __builtin_amdgcn_wmma_f32_16x16x4_f32(
          false, a, false, b, (short)0, acc[t], false, false);
    }
  }

  // ---- activation + store ----
#pragma unroll
  for (int t = 0; t < 8; t++)
#pragma unroll
    for (int v = 0; v < 8; v++) {
      float y = acc[t][v];
      if (do_relu) y = fmaxf(y, 0.f);
      Yout[(size_t)(row0 + mbase + v) * NDIM + t * 16 + n] = y;
    }
}

extern "C" void kernel_launch(void* const* d_in, const int* in_sizes, int n_in,
                              void* d_out, int out_size, void* d_ws, size_t ws_size,
                              hipStream_t stream) {
  const int*   entity     = (const int*)d_in[0];
  const int*   edge_index = (const int*)d_in[1];
  const int*   edge_type  = (const int*)d_in[2];
  /* d_in[3] = edge_norm (unused by the reference math) */
  const float* emb   = (const float*)d_in[4];
  const float* W1    = (const float*)d_in[5];
  const float* root1 = (const float*)d_in[6];
  const float* W2    = (const float*)d_in[7];
  const float* root2 = (const float*)d_in[8];

  const int N  = in_sizes[0];
  const int E  = in_sizes[2];
  const int NR = N * RNUM;
  const int* src = edge_index;
  const int* dst = edge_index + E;

  // ---- workspace layout (~126 MB) ----
  size_t nd = (size_t)N * NDIM;
  float* X      = (float*)d_ws;        // [N,128]
  float* Y      = X + nd;              // [N,128]
  float* inv    = Y + nd;              // [N]
  int*   cnt    = (int*)(inv + N);     // [N*16]
  int*   cursor = cnt + NR;            // [N*16]  (adjacent to cnt: zeroed together)
  int*   off    = cursor + NR;         // [N*16]
  int*   srcS   = off + NR;            // [E]
  int*   s1     = srcS + E;            // [8192]
  int*   s1x    = s1 + 8192;           // [8192]
  int*   s2     = s1x + 8192;          // [256]
  int*   s2x    = s2 + 256;            // [256]

  const int TB = 256;
  int gatherBlocks = (int)(((size_t)N * 32 + TB - 1) / TB);
  int edgeBlocks   = (E + TB - 1) / TB;
  int NB1 = (NR + 255) / 256;                  // scan level-1 blocks (6250)
  int NB2 = (NB1 + 255) / 256;                 // level-2 blocks (25)
  long z4 = (long)(2 * NR) / 4;                // cnt + cursor as int4
  int tiles = (N + 15) / 16;
  int layerBlocks = (tiles + 3) / 4;           // 4 waves per 128-thread block

  // ---- one-time prep ----
  gather_rows<<<gatherBlocks, TB, 0, stream>>>(entity, emb, X, N);
  zero_i4<<<(int)((z4 + TB - 1) / TB), TB, 0, stream>>>((int4*)cnt, z4);
  histo<<<edgeBlocks, TB, 0, stream>>>(dst, edge_type, cnt, E);
  scan_block<<<NB1, 256, 0, stream>>>(cnt, off, s1, NR);
  scan_block<<<NB2, 256, 0, stream>>>(s1, s1x, s2, NB1);
  scan_block<<<1, 256, 0, stream>>>(s2, s2x, nullptr, NB2);
  scan_fix<<<NB1, 256, 0, stream>>>(off, s1x, s2x, NR);
  place_edges<<<edgeBlocks, TB, 0, stream>>>(src, dst, edge_type, off, cursor, srcS, E);
  make_inv<<<(N + TB - 1) / TB, TB, 0, stream>>>(cnt, inv, N);

  // ---- two fused layers ----
  rgcn_layer<<<layerBlocks, 128, 0, stream>>>(X, srcS, off, cnt, W1, root1, inv,
                                              Y, N, /*relu=*/1);
  rgcn_layer<<<layerBlocks, 128, 0, stream>>>(Y, srcS, off, cnt, W2, root2, inv,
                                              (float*)d_out, N, /*relu=*/0);
}